// PostProcessor_68556267978758
// MI455X (gfx1250) — compile-verified
//
#include <hip/hip_runtime.h>
#include <stdint.h>
#include <math.h>

// Problem constants (from reference)
constexpr int kB  = 32;
constexpr int kN  = 1000;
constexpr int kC  = 105;
constexpr int kC1 = 104;                 // classes 1..104 (skip background)
constexpr int kDet = 100;
constexpr float kScoreTh = 0.05f;
constexpr float kNmsTh   = 0.5f;
constexpr float kImgW = 1333.0f;
constexpr float kImgH = 800.0f;
constexpr float kBoxClip = 4.135166556742356f;   // log(1000/16)
#define NEG_INF (-__builtin_inff())

constexpr int kM  = 1024;                // padded NMS problem size
constexpr int kTN = 256;                 // NMS block (8 wave32)
constexpr int kTK = 1024;                // topk block (32 wave32)
constexpr int kP  = kC1 * kN;            // 104000 candidates per image

// ---- CDNA5 async global->LDS staging (ASYNCcnt path), inline asm for
// toolchain portability (bypasses the differing clang builtin arities).
static __device__ __forceinline__ void async_b128_to_lds(const void* g, const void* lds) {
  asm volatile("global_load_async_to_lds_b128 %0, %1, off"
               :: "v"((unsigned)(unsigned long long)(uintptr_t)lds),
                  "v"((unsigned long long)(uintptr_t)g)
               : "memory");
}
static __device__ __forceinline__ void async_wait_all() {
  asm volatile("s_wait_asynccnt 0" ::: "memory");
}

// ------------------------------------------------------------------
// Kernel 1: softmax over C=105, one wave32 per row, output transposed
// to prob[(b*kC1 + (c-1))*kN + n]
// ------------------------------------------------------------------
__global__ __launch_bounds__(256) void softmax_k(const float* __restrict__ logits,
                                                 float* __restrict__ prob) {
  const int wave = threadIdx.x >> 5;
  const int lane = threadIdx.x & 31;
  const int row  = blockIdx.x * 8 + wave;          // < kB*kN (exact)
  const float* src = logits + (size_t)row * kC;

  float v[4];
  float m = NEG_INF;
#pragma unroll
  for (int k = 0; k < 4; ++k) {
    const int c = lane + 32 * k;
    v[k] = (c < kC) ? src[c] : NEG_INF;
    m = fmaxf(m, v[k]);
  }
#pragma unroll
  for (int off = 16; off; off >>= 1) m = fmaxf(m, __shfl_xor(m, off, 32));

  float e[4];
  float s = 0.0f;
#pragma unroll
  for (int k = 0; k < 4; ++k) {
    const int c = lane + 32 * k;
    e[k] = (c < kC) ? expf(v[k] - m) : 0.0f;
    s += e[k];
  }
#pragma unroll
  for (int off = 16; off; off >>= 1) s += __shfl_xor(s, off, 32);
  const float inv = 1.0f / s;

  const int b = row / kN, n = row % kN;
#pragma unroll
  for (int k = 0; k < 4; ++k) {
    const int c = lane + 32 * k;
    if (c >= 1 && c < kC)
      prob[((size_t)b * kC1 + (c - 1)) * kN + n] = e[k] * inv;
  }
}

// ------------------------------------------------------------------
// Kernel 2: decode + clip, output transposed to (b, class, n, 4)
// ------------------------------------------------------------------
__global__ __launch_bounds__(256) void decode_k(const float4* __restrict__ reg,
                                                const float4* __restrict__ prop,
                                                float4* __restrict__ dec) {
  const int gid = blockIdx.x * blockDim.x + threadIdx.x;  // < kB*kN*kC1 (exact)
  const int j1 = gid % kC1;
  const int t  = gid / kC1;                               // b*kN + n

  const float4 r = reg[(size_t)t * kC + (j1 + 1)];
  const float4 p = prop[t];

  const float w  = p.z - p.x + 1.0f, h = p.w - p.y + 1.0f;
  const float cx = p.x + 0.5f * w,  cy = p.y + 0.5f * h;
  const float dx = r.x / 10.0f, dy = r.y / 10.0f;
  const float dw = fminf(r.z / 5.0f, kBoxClip);
  const float dh = fminf(r.w / 5.0f, kBoxClip);
  const float pcx = dx * w + cx, pcy = dy * h + cy;
  const float pw = expf(dw) * w, ph = expf(dh) * h;

  float4 o;
  o.x = fminf(fmaxf(pcx - 0.5f * pw,        0.0f), kImgW - 1.0f);
  o.y = fminf(fmaxf(pcy - 0.5f * ph,        0.0f), kImgH - 1.0f);
  o.z = fminf(fmaxf(pcx + 0.5f * pw - 1.0f, 0.0f), kImgW - 1.0f);
  o.w = fminf(fmaxf(pcy + 0.5f * ph - 1.0f, 0.0f), kImgH - 1.0f);

  const int n = t % kN, b = t / kN;
  dec[((size_t)b * kC1 + j1) * kN + n] = o;
}

// ------------------------------------------------------------------
// Kernel 3: per-(image,class) NMS. One block per problem. LDS-resident:
// async-staged boxes+scores, stable bitonic sort, greedy suppression over
// the finite-score prefix only, barriers only after suppression writes.
// Outputs sorted boxes (all 1000) and scores (kept ? s : -inf).
// ------------------------------------------------------------------
__global__ __launch_bounds__(kTN) void nms_k(const float* __restrict__ prob,
                                             const float4* __restrict__ dec,
                                             float* __restrict__ oScore,
                                             float4* __restrict__ oBox) {
  __shared__ float4 sBox[kM];
  __shared__ __align__(16) float sScore[kM];
  __shared__ int sIdx[kM];
  __shared__ float sx1[kM], sy1[kM], sx2[kM], sy2[kM], sA[kM];
  __shared__ unsigned char sSupp[kM];
  __shared__ unsigned char sKeep[kM];
  __shared__ int sCnt;

  const int pair = blockIdx.x;                 // b*kC1 + j1
  const int tid  = threadIdx.x;
  const size_t base = (size_t)pair * kN;
  const float*  gS = prob + base;              // contiguous 4000B
  const float4* gB = dec + base;               // contiguous 16000B

  if (tid == 0) sCnt = 0;

  // Async stage: 250 x B128 for scores, 1000 x B128 for boxes.
  for (int q = tid; q < kN / 4; q += kTN)
    async_b128_to_lds(gS + 4 * q, &sScore[4 * q]);
  for (int n = tid; n < kN; n += kTN)
    async_b128_to_lds(gB + n, &sBox[n]);
  async_wait_all();
  __syncthreads();

  // Threshold to -inf (reference does this BEFORE the sort), init pads,
  // and count finite (above-threshold) entries.
  int myCnt = 0;
  for (int i = tid; i < kM; i += kTN) {
    const float s = (i < kN) ? sScore[i] : NEG_INF;
    const bool ok = (s > kScoreTh);
    sScore[i] = ok ? s : NEG_INF;
    myCnt += ok ? 1 : 0;
    sIdx[i]  = i;
    sSupp[i] = 0;
    sKeep[i] = 0;
  }
  atomicAdd(&sCnt, myCnt);

  // Stable bitonic sort: descending score, ascending original index on ties
  // (keys are unique since indices are unique => equals jnp stable argsort).
  for (int k = 2; k <= kM; k <<= 1) {
    for (int j2 = k >> 1; j2 > 0; j2 >>= 1) {
      __syncthreads();
      for (int i = tid; i < kM; i += kTN) {
        const int ix = i ^ j2;
        if (ix > i) {
          const float va = sScore[i], vb = sScore[ix];
          const int   ia = sIdx[i],   ib = sIdx[ix];
          const bool aFirst = (va > vb) || ((va == vb) && (ia < ib));
          const bool up = ((i & k) == 0);
          if (up ? !aFirst : aFirst) {
            sScore[i] = vb; sScore[ix] = va;
            sIdx[i]   = ib; sIdx[ix]   = ia;
          }
        }
      }
    }
  }
  __syncthreads();

  // After sort, positions [0, limit) hold exactly the finite scores.
  const int limit = sCnt;

  // Gather sorted coordinates + areas (legacy +1 convention).
  for (int i = tid; i < kM; i += kTN) {
    const int o = sIdx[i];
    const float4 bx = (o < kN) ? sBox[o] : make_float4(0.f, 0.f, 0.f, 0.f);
    sx1[i] = bx.x; sy1[i] = bx.y; sx2[i] = bx.z; sy2[i] = bx.w;
    sA[i] = (bx.z - bx.x + 1.0f) * (bx.w - bx.y + 1.0f);
  }
  __syncthreads();

  // Greedy scan over the finite prefix. All threads evaluate the same
  // predicate from LDS data that only changes across barriers, so the
  // conditional barrier is uniform across the block.
  for (int i = 0; i < limit; ++i) {
    const bool keep = (sSupp[i] == 0);
    if (tid == 0) sKeep[i] = keep ? 1 : 0;
    if (keep) {
      const float x1 = sx1[i], y1 = sy1[i], x2 = sx2[i], y2 = sy2[i], ai = sA[i];
      for (int e = tid; e < limit; e += kTN) {
        if (e > i) {
          const float xx1 = fmaxf(x1, sx1[e]), yy1 = fmaxf(y1, sy1[e]);
          const float xx2 = fminf(x2, sx2[e]), yy2 = fminf(y2, sy2[e]);
          const float ww = fmaxf(xx2 - xx1 + 1.0f, 0.0f);
          const float hh = fmaxf(yy2 - yy1 + 1.0f, 0.0f);
          const float inter = ww * hh;
          const float iou = inter / (ai + sA[e] - inter);
          if (iou > kNmsTh) sSupp[e] = 1;
        }
      }
      __syncthreads();
    }
  }
  __syncthreads();

  // Emit (pads at positions >= 1000 by construction).
  for (int i = tid; i < kN; i += kTN) {
    oScore[base + i] = sKeep[i] ? sScore[i] : NEG_INF;
    oBox[base + i] = make_float4(sx1[i], sy1[i], sx2[i], sy2[i]);
  }
}

// ------------------------------------------------------------------
// Kernel 4: per-image top-100 (lowest-index tie-break, matching lax.top_k)
// Wave32 shuffle reduction + single 32-entry LDS stage.
// ------------------------------------------------------------------
__global__ __launch_bounds__(kTK) void topk_k(const float* __restrict__ sc,
                                              const float4* __restrict__ bx,
                                              float* __restrict__ det,
                                              float* __restrict__ lab,
                                              float* __restrict__ val) {
  __shared__ float rv[32];
  __shared__ int   ri[32];
  __shared__ unsigned sTaken[(kP + 31) / 32];

  const int b = blockIdx.x, tid = threadIdx.x;
  const int lane = tid & 31, wid = tid >> 5;
  const float* s = sc + (size_t)b * kP;

  for (int i = tid; i < (kP + 31) / 32; i += kTK) sTaken[i] = 0u;
  __syncthreads();

  for (int r = 0; r < kDet; ++r) {
    float bv = NEG_INF;
    int   bi = 0x7FFFFFFF;
    for (int f = tid; f < kP; f += kTK) {
      if ((sTaken[f >> 5] >> (f & 31)) & 1u) continue;
      const float v = s[f];
      if (v > bv || (v == bv && f < bi)) { bv = v; bi = f; }
    }
    // wave32 reduction
#pragma unroll
    for (int off = 16; off; off >>= 1) {
      const float v2 = __shfl_xor(bv, off, 32);
      const int   i2 = __shfl_xor(bi, off, 32);
      if (v2 > bv || (v2 == bv && i2 < bi)) { bv = v2; bi = i2; }
    }
    if (lane == 0) { rv[wid] = bv; ri[wid] = bi; }
    __syncthreads();
    if (wid == 0) {
      bv = rv[lane]; bi = ri[lane];
#pragma unroll
      for (int off = 16; off; off >>= 1) {
        const float v2 = __shfl_xor(bv, off, 32);
        const int   i2 = __shfl_xor(bi, off, 32);
        if (v2 > bv || (v2 == bv && i2 < bi)) { bv = v2; bi = i2; }
      }
      if (lane == 0) {
        const int f = bi;
        sTaken[f >> 5] |= (1u << (f & 31));
        const float4 bb = bx[(size_t)b * kP + f];
        const bool fin = (bv != NEG_INF);
        float* d = det + ((size_t)b * kDet + r) * 5;
        d[0] = bb.x; d[1] = bb.y; d[2] = bb.z; d[3] = bb.w;
        d[4] = fin ? bv : 0.0f;
        lab[b * kDet + r] = (float)(f / kN + 1);
        val[b * kDet + r] = fin ? 1.0f : 0.0f;
      }
    }
    __syncthreads();
  }
}

// ------------------------------------------------------------------
extern "C" void kernel_launch(void* const* d_in, const int* in_sizes, int n_in,
                              void* d_out, int out_size, void* d_ws, size_t ws_size,
                              hipStream_t stream) {
  const float* logits = (const float*)d_in[0];   // (B,N,C)
  const float* reg    = (const float*)d_in[1];   // (B,N,4C)
  const float* prop   = (const float*)d_in[2];   // (B,N,4)

  float* ws = (float*)d_ws;
  const size_t nPC = (size_t)kB * kC1 * kN;      // 3,328,000
  float* prob = ws;                              // nPC floats
  float* dec  = prob + nPC;                      // nPC float4s
  float* ssc  = dec + nPC * 4;                   // nPC floats
  float* sbx  = ssc + nPC;                       // nPC float4s

  float* det = (float*)d_out;                    // B*100*5
  float* lab = det + kB * kDet * 5;              // B*100
  float* val = lab + kB * kDet;                  // B*100

  softmax_k<<<dim3(kB * kN / 8), dim3(256), 0, stream>>>(logits, prob);
  decode_k<<<dim3((kB * kN * kC1) / 256), dim3(256), 0, stream>>>(
      (const float4*)reg, (const float4*)prop, (float4*)dec);
  nms_k<<<dim3(kB * kC1), dim3(kTN), 0, stream>>>(
      prob, (const float4*)dec, ssc, (float4*)sbx);
  topk_k<<<dim3(kB), dim3(kTK), 0, stream>>>(
      ssc, (const float4*)sbx, det, lab, val);
}